// SparseFF_54193897341184
// MI455X (gfx1250) — compile-verified
//
#include <hip/hip_runtime.h>

typedef __attribute__((ext_vector_type(2))) float v2f;
typedef __attribute__((ext_vector_type(8))) float v8f;

#define NTOK   2048   // BATCH*SEQ
#define DM     1024   // d_model
#define DLR    64     // d_lowrank
#define ND1    64     // d1 (blocks)
#define ND2    64     // d2 (choices per block)

// ---------------------------------------------------------------------------
// Router: lr = x@m1 ; logits = lr@m2 + mb ; qm = argmax_d2(logits)
// One workgroup handles 16 tokens. 4 waves (128 threads).
// WMMA f32 16x16x4 per ISA layouts:
//   A 16x4:  lanes 0-15: M=lane, VGPR0=K0,VGPR1=K1 ; lanes 16-31: K2,K3
//   B 4x16:  VGPR0: lanes0-15=(K0,N=lane), lanes16-31=(K2,N=lane-16); VGPR1: K1/K3
//   C 16x16: VGPR v: lanes0-15=(M=v,N=lane), lanes16-31=(M=v+8,N=lane-16)
// ---------------------------------------------------------------------------
__global__ __launch_bounds__(128) void sff_router_kernel(
    const float* __restrict__ x, const float* __restrict__ m1,
    const float* __restrict__ m2, const float* __restrict__ mb,
    int* __restrict__ qm)
{
  __shared__ float xs[16 * DM];        // 64 KB: x tile, 16 tokens
  __shared__ float lrs[16 * DLR];      // 4 KB : lr tile
  __shared__ float lg[4][16 * ND2];    // 16 KB: per-wave logits scratch

  const int tid  = threadIdx.x;
  const int wave = tid >> 5;
  const int lane = tid & 31;
  const int half = lane >> 4;          // 0: lanes 0-15, 1: lanes 16-31
  const int l16  = lane & 15;
  const int tok0 = blockIdx.x * 16;

  // cooperative load of 16 x rows (16384 floats) as float4
  {
    const float4* xg  = (const float4*)(x + (size_t)tok0 * DM);
    float4*       xs4 = (float4*)xs;
    for (int i = tid; i < 16 * DM / 4; i += 128) xs4[i] = xg[i];
  }
  __syncthreads();

  // ---- Stage 1: lr[16 x 64]; wave w owns columns w*16 .. w*16+15 ----
  {
    v8f acc = {};
    const int ncol = wave * 16 + l16;
    for (int k0 = 0; k0 < DM; k0 += 4) {
      v2f a, b;
      const float* xr = xs + l16 * DM + k0 + half * 2;
      a.x = xr[0];
      a.y = xr[1];
      const float* mr = m1 + (size_t)(k0 + half * 2) * DLR + ncol;
      b.x = mr[0];
      b.y = mr[DLR];
      acc = __builtin_amdgcn_wmma_f32_16x16x4_f32(false, a, false, b,
                                                  (short)0, acc, false, false);
    }
    for (int v = 0; v < 8; ++v) lrs[(v + half * 8) * DLR + wave * 16 + l16] = acc[v];
  }
  __syncthreads();

  // ---- Stage 2: per block j: logits[16 x 64] = lr @ m2[:,j,:] + mb[j,:] ----
  float* mylg = lg[wave];
  for (int j = wave; j < ND1; j += 4) {     // 16 uniform iterations per wave
    for (int nt = 0; nt < 4; ++nt) {
      v8f acc = {};
      const int ncol = nt * 16 + l16;
      for (int k0 = 0; k0 < DLR; k0 += 4) {
        v2f a, b;
        const float* lrp = lrs + l16 * DLR + k0 + half * 2;
        a.x = lrp[0];
        a.y = lrp[1];
        const float* m2p = m2 + (size_t)(k0 + half * 2) * (ND1 * ND2) + j * ND2 + ncol;
        b.x = m2p[0];
        b.y = m2p[ND1 * ND2];
        acc = __builtin_amdgcn_wmma_f32_16x16x4_f32(false, a, false, b,
                                                    (short)0, acc, false, false);
      }
      const float mbv = mb[j * ND2 + ncol];
      for (int v = 0; v < 8; ++v)
        mylg[(v + half * 8) * ND2 + ncol] = acc[v] + mbv;
    }
    __syncthreads();                        // uniform across all 4 waves
    if (lane < 16) {
      const float* row = mylg + lane * ND2;
      float best = row[0];
      int   bi   = 0;
      for (int m = 1; m < ND2; ++m) {       // strict '>' keeps FIRST max (jnp.argmax)
        float v = row[m];
        if (v > best) { best = v; bi = m; }
      }
      qm[(size_t)(tok0 + lane) * ND1 + j] = bi;
    }
    __syncthreads();
  }
}

// ---------------------------------------------------------------------------
// Transpose w1[k, j, m] (DM x ND1 x ND2) -> w1t[j, m, k] so the per-token
// gathered column becomes a contiguous 4KB row (coalesced reads).
// grid = (64 j, 16 k-chunks), 256 threads, 64x64 LDS tile.
// ---------------------------------------------------------------------------
__global__ __launch_bounds__(256) void sff_transpose_w1_kernel(
    const float* __restrict__ w1, float* __restrict__ w1t)
{
  __shared__ float tile[64][65];
  const int j  = blockIdx.x;
  const int k0 = blockIdx.y * 64;
  const int t  = threadIdx.x;
  for (int i = t; i < 64 * 64; i += 256) {
    int r = i >> 6, c = i & 63;             // r: k offset, c: m  (c contiguous -> coalesced)
    tile[r][c] = w1[(size_t)(k0 + r) * (ND1 * ND2) + (size_t)j * ND2 + c];
  }
  __syncthreads();
  for (int i = t; i < 64 * 64; i += 256) {
    int m = i >> 6, kk = i & 63;            // kk contiguous -> coalesced store
    w1t[(size_t)j * (ND2 * DM) + (size_t)m * DM + (size_t)(k0 + kk)] = tile[kk][m];
  }
}

// ---------------------------------------------------------------------------
// Main path: one workgroup per token (256 threads = 8 waves).
//   mid[j] = x . w1col(j, qm[j]);  r = relu(mid);  out = sum_j r[j]*w2[qm[j],j,:] + b2
// ---------------------------------------------------------------------------
__global__ __launch_bounds__(256) void sff_main_kernel(
    const float* __restrict__ x,  const float* __restrict__ w1,
    const float* __restrict__ w1t, const int use_t,
    const float* __restrict__ w2, const float* __restrict__ b2,
    const int* __restrict__ qm,   float* __restrict__ out)
{
  __shared__ float xsm[DM];
  __shared__ int   qs[ND1];
  __shared__ float rs[ND1];

  const int b    = blockIdx.x;
  const int t    = threadIdx.x;
  const int wave = t >> 5;
  const int lane = t & 31;

  ((float4*)xsm)[t] = ((const float4*)(x + (size_t)b * DM))[t];
  if (t < ND1) qs[t] = qm[(size_t)b * ND1 + t];
  __syncthreads();

  // mid + relu: wave w handles blocks j = w*8 .. w*8+7
  for (int j = wave * 8; j < wave * 8 + 8; ++j) {
    const int q = qs[j];
    float p = 0.f;
    if (use_t) {
      const float* wr = w1t + (size_t)j * (ND2 * DM) + (size_t)q * DM;
      for (int k = lane; k < DM; k += 32) p += xsm[k] * wr[k];          // coalesced
    } else {
      const float* wr = w1 + (size_t)j * ND2 + q;
      for (int k = lane; k < DM; k += 32) p += xsm[k] * wr[(size_t)k * (ND1 * ND2)];
    }
    for (int off = 16; off > 0; off >>= 1) p += __shfl_down(p, off, 32); // wave32 reduce
    if (lane == 0) rs[j] = p > 0.f ? p : 0.f;
  }
  __syncthreads();

  // out[b, t*4 .. t*4+3]
  float4 acc = ((const float4*)b2)[t];
  for (int j = 0; j < ND1; ++j) {
    const float  r = rs[j];
    const float4 w = ((const float4*)(w2 + (size_t)qs[j] * (ND1 * DM) + (size_t)j * DM))[t];
    acc.x += r * w.x; acc.y += r * w.y; acc.z += r * w.z; acc.w += r * w.w;
  }
  ((float4*)(out + (size_t)b * DM))[t] = acc;
}

extern "C" void kernel_launch(void* const* d_in, const int* in_sizes, int n_in,
                              void* d_out, int out_size, void* d_ws, size_t ws_size,
                              hipStream_t stream) {
  const float* x  = (const float*)d_in[0];
  const float* m1 = (const float*)d_in[1];
  const float* m2 = (const float*)d_in[2];
  const float* mb = (const float*)d_in[3];
  const float* w1 = (const float*)d_in[4];
  const float* w2 = (const float*)d_in[5];
  const float* b2 = (const float*)d_in[6];
  float* out = (float*)d_out;

  int*   qm       = (int*)d_ws;
  size_t qm_bytes = (size_t)NTOK * ND1 * sizeof(int);                 // 512 KB
  float* w1t      = (float*)((char*)d_ws + qm_bytes);
  size_t need     = qm_bytes + (size_t)ND1 * ND2 * DM * sizeof(float); // +16 MB
  const int use_t = (ws_size >= need) ? 1 : 0;

  sff_router_kernel<<<NTOK / 16, 128, 0, stream>>>(x, m1, m2, mb, qm);
  if (use_t)
    sff_transpose_w1_kernel<<<dim3(ND1, DM / 64), 256, 0, stream>>>(w1, w1t);
  sff_main_kernel<<<NTOK, 256, 0, stream>>>(x, w1, use_t ? w1t : nullptr, use_t,
                                            w2, b2, qm, out);
}